// ATSS_25675314495726
// MI455X (gfx1250) — compile-verified
//
#include <hip/hip_runtime.h>
#include <stdint.h>

typedef float v2f __attribute__((ext_vector_type(2)));
typedef float v8f __attribute__((ext_vector_type(8)));

#define NB     16        // batches
#define NPRED  16384
#define NGT    128
#define KSEL   64
#define GTILE  16        // gts per block (== waves per block)
#define CHUNK  512       // preds per streamed chunk
#define NCHUNK (NPRED / CHUNK)

// ---- wave32 cross-lane helpers (ds_bpermute: index bits [6:2] select lane) ----
__device__ __forceinline__ unsigned bcast_u32(unsigned v, int src) {
  return (unsigned)__builtin_amdgcn_ds_bpermute(src << 2, (int)v);
}
__device__ __forceinline__ float bcast_f32(float v, int src) {
  return __int_as_float(__builtin_amdgcn_ds_bpermute(src << 2, __float_as_int(v)));
}
__device__ __forceinline__ uint64_t bcast_u64(uint64_t v, int src) {
  unsigned lo = bcast_u32((unsigned)v, src);
  unsigned hi = bcast_u32((unsigned)(v >> 32), src);
  return ((uint64_t)hi << 32) | (uint64_t)lo;
}
__device__ __forceinline__ float wave_sum64(float v, int lane) {
  // sums the 32 per-lane values (each lane already holds a partial of 2 elems)
  #pragma unroll
  for (int off = 16; off > 0; off >>= 1) v += bcast_f32(v, lane ^ off);
  return v;
}

__global__ __launch_bounds__(512)
void ATSS_kernel(const float* __restrict__ pred,
                 const float* __restrict__ gt,
                 float* __restrict__ out) {
  __shared__ unsigned chunkDist[GTILE][CHUNK];   // 32 KB

  const int lane = (int)(threadIdx.x & 31u);
  const int wave = (int)(threadIdx.x >> 5);      // 0..15, owns gt 'wave' of the tile
  const int b    = (int)(blockIdx.x >> 3);       // 8 gt-tiles per batch
  const int gt0  = ((int)blockIdx.x & 7) * GTILE;

  // ---- load this tile's 16 gt boxes into lanes 0..15 of every wave ----
  float gcx = 0.f, gcy = 0.f, gwd = 0.f, ght = 0.f;
  if (lane < 16) {
    const float* gp = gt + ((size_t)b * NGT + (size_t)(gt0 + lane)) * 4;
    gcx = gp[0]; gcy = gp[1]; gwd = gp[2]; ght = gp[3];
  }
  const float gxs = gcx - 0.5f, gys = gcy - 0.5f;   // centered coords (cancellation)
  v2f amat;                                         // A row m=lane: K0=-2gx, K1=-2gy (K2=K3=0)
  amat.x = (lane < 16) ? (-2.0f * gxs) : 0.0f;
  amat.y = (lane < 16) ? (-2.0f * gys) : 0.0f;
  const float gn2 = gxs * gxs + gys * gys;
  const int hi8 = (lane >> 4) << 3;                 // 0 for lanes 0-15, 8 for lanes 16-31
  float gnC[8];
  #pragma unroll
  for (int r = 0; r < 8; ++r) gnC[r] = bcast_f32(gn2, r + hi8);  // |g|^2 for row m=r+hi8

  // ---- register-resident sorted top-64: slot 'lane' in a0, slot 'lane+32' in a1 ----
  uint64_t a0 = ~0ull, a1 = ~0ull, curMax = ~0ull;

  for (int c = 0; c < NCHUNK; ++c) {
    // ===== Phase 1: distances via v_wmma_f32_16x16x4_f32 (2 tiles / wave) =====
    #pragma unroll
    for (int t = 0; t < 2; ++t) {
      const int nb0 = (wave * 2 + t) * 16;          // pred offset within chunk
      const int pb  = c * CHUNK + nb0;
      float px = 0.f, py = 0.f;
      if (lane < 16) {
        const float* pp = pred + ((size_t)b * NPRED + (size_t)(pb + lane)) * 4;
        px = pp[0] - 0.5f; py = pp[1] - 0.5f;
      }
      v2f bmat;                                     // B col n=lane: K0=px, K1=py (K2=K3=0)
      bmat.x = (lane < 16) ? px : 0.0f;
      bmat.y = (lane < 16) ? py : 0.0f;
      const float pn2 = px * px + py * py;
      const float pn  = bcast_f32(pn2, lane & 15);  // |p|^2 for col n = lane%16
      v8f cmat;
      #pragma unroll
      for (int r = 0; r < 8; ++r) cmat[r] = gnC[r] + pn;   // C[m][n] = |g|^2 + |p|^2
      // D = A*B + C  =>  squared distance matrix tile (16 gts x 16 preds)
      v8f d2 = __builtin_amdgcn_wmma_f32_16x16x4_f32(
          false, amat, false, bmat, (short)0, cmat, false, false);
      const int n = nb0 + (lane & 15);
      #pragma unroll
      for (int r = 0; r < 8; ++r) {
        float dv = sqrtf(fmaxf(d2[r], 0.0f));       // sqrt: match reference sort key
        chunkDist[r + hi8][n] = __float_as_uint(dv);
      }
    }
    __syncthreads();

    // ===== Phase 2: wave w merges chunk row w into its top-64 =====
    for (int base = 0; base < CHUNK; base += 32) {
      const unsigned db = chunkDist[wave][base + lane];
      const uint64_t key = ((uint64_t)db << 32) | (uint64_t)(unsigned)(c * CHUNK + base + lane);
      unsigned qm = __builtin_amdgcn_ballot_w32(key < curMax);
      while (qm) {
        const int s = __ffs(qm) - 1; qm &= qm - 1;
        const uint64_t k = bcast_u64(key, s);       // uniform candidate
        if (k >= curMax) continue;                  // stale after earlier inserts (uniform)
        const int pos = (int)(__builtin_popcount(__builtin_amdgcn_ballot_w32(a0 < k)) +
                              __builtin_popcount(__builtin_amdgcn_ballot_w32(a1 < k)));
        // shift network: new[j] = j<pos ? old[j] : (j==pos ? k : old[j-1])
        uint64_t am0 = bcast_u64(a0, lane - 1);     // old[lane-1]   (lane0 value unused)
        uint64_t top0 = bcast_u64(a0, 31);          // old[31]
        uint64_t am1 = bcast_u64(a1, lane - 1);     // old[lane+31]  for lane>=1
        if (lane == 0) am1 = top0;
        const int j1 = lane + 32;
        a0 = (lane < pos) ? a0 : ((lane == pos) ? k : am0);
        a1 = (j1   < pos) ? a1 : ((j1   == pos) ? k : am1);
        curMax = bcast_u64(a1, 31);
      }
    }
    __syncthreads();
  }

  // ===== Phase 3: IoU, threshold, mask, outputs (wave w -> gt gt0+w) =====
  const float bgx = bcast_f32(gcx, wave), bgy = bcast_f32(gcy, wave);
  const float bgw = bcast_f32(gwd, wave), bgh = bcast_f32(ght, wave);
  const float gx1 = bgx - 0.5f * bgw, gy1 = bgy - 0.5f * bgh;
  const float gx2 = bgx + 0.5f * bgw, gy2 = bgy + 0.5f * bgh;
  const float areaG = (gx2 - gx1) * (gy2 - gy1);

  float iou[2], insf[2], idxf[2];
  const uint64_t keys[2] = { a0, a1 };
  #pragma unroll
  for (int e = 0; e < 2; ++e) {
    const unsigned p = (unsigned)keys[e];
    const float* pp = pred + ((size_t)b * NPRED + p) * 4;
    const float cx = pp[0], cy = pp[1], w = pp[2], h = pp[3];
    const float px1 = cx - 0.5f * w, py1 = cy - 0.5f * h;
    const float px2 = cx + 0.5f * w, py2 = cy + 0.5f * h;
    const float ltx = fmaxf(gx1, px1), lty = fmaxf(gy1, py1);
    const float rbx = fminf(gx2, px2), rby = fminf(gy2, py2);
    const float iw = fmaxf(rbx - ltx, 0.0f), ih = fmaxf(rby - lty, 0.0f);
    const float inter = iw * ih;
    const float areaP = (px2 - px1) * (py2 - py1);
    iou[e]  = inter / (areaG + areaP - inter);
    insf[e] = ((gx1 <= cx) && (cx <= gx2) && (gy1 <= cy) && (cy <= gy2)) ? 1.0f : 0.0f;
    idxf[e] = (float)p;
  }

  const float mean = wave_sum64(iou[0] + iou[1], lane) * (1.0f / 64.0f);
  const float d0 = iou[0] - mean, d1 = iou[1] - mean;
  const float ss = wave_sum64(d0 * d0 + d1 * d1, lane);
  const float thr = mean + sqrtf(ss / 63.0f);       // std with ddof=1

  const size_t TOT   = (size_t)NB * NGT * KSEL;     // 131072 per output tensor
  const size_t gbase = ((size_t)b * NGT + (size_t)(gt0 + wave)) * KSEL;
  #pragma unroll
  for (int e = 0; e < 2; ++e) {
    const size_t kk = gbase + (size_t)(lane + 32 * e);
    out[kk]           = iou[e];
    out[TOT + kk]     = ((iou[e] >= thr) && (insf[e] != 0.0f)) ? 1.0f : 0.0f;
    out[2 * TOT + kk] = idxf[e];
  }
}

extern "C" void kernel_launch(void* const* d_in, const int* in_sizes, int n_in,
                              void* d_out, int out_size, void* d_ws, size_t ws_size,
                              hipStream_t stream) {
  (void)in_sizes; (void)n_in; (void)out_size; (void)d_ws; (void)ws_size;
  const float* pred = (const float*)d_in[0];   // (16, 16384, 4) f32
  const float* gtb  = (const float*)d_in[1];   // (16, 128, 4)   f32
  float* out = (float*)d_out;                  // [ious | mask | k_idx] flat f32
  ATSS_kernel<<<dim3(NB * (NGT / GTILE)), dim3(GTILE * 32), 0, stream>>>(pred, gtb, out);
}